// Autoencoder_8950711845588
// MI455X (gfx1250) — compile-verified
//
#include <hip/hip_runtime.h>
#include <math.h>

typedef __attribute__((ext_vector_type(2))) float v2f;
typedef __attribute__((ext_vector_type(8))) float v8f;

#define BB 8
#define PP 4096
#define NS1 820
#define NS2 205
#define KN 64
#define EPS_BN 1e-5f
#define RCAP 1024

// ---------------------------------------------------------------- FPS
__global__ void pn2_fps_kernel(const float* __restrict__ pos, int* __restrict__ out,
                               int P, int S) {
  int b = blockIdx.x;
  const float* p = pos + (size_t)b * P * 3;
  int* o = out + (size_t)b * S;
  __shared__ float sval[256];
  __shared__ int   sidx[256];
  __shared__ float lastp[3];
  __shared__ int   lasti;
  int t = threadIdx.x;
  float dmin[16];
  int cpt = (P + 255) >> 8;
  for (int i = 0; i < cpt; ++i) dmin[i] = INFINITY;
  if (t == 0) { o[0] = 0; lasti = 0; }
  __syncthreads();
  for (int s = 1; s < S; ++s) {
    if (t < 3) lastp[t] = p[(size_t)lasti * 3 + t];
    __syncthreads();
    float bx = lastp[0], by = lastp[1], bz = lastp[2];
    float bestv = -INFINITY; int besti = 0x7fffffff;
    for (int i = 0; i < cpt; ++i) {
      int pi = t + (i << 8);
      if (pi < P) {
        float dx = p[pi*3+0]-bx, dy = p[pi*3+1]-by, dz = p[pi*3+2]-bz;
        float d = dx*dx + dy*dy + dz*dz;
        float dm = fminf(dmin[i], d);
        dmin[i] = dm;
        if (dm > bestv || (dm == bestv && pi < besti)) { bestv = dm; besti = pi; }
      }
    }
    sval[t] = bestv; sidx[t] = besti;
    __syncthreads();
    for (int off = 128; off > 0; off >>= 1) {
      if (t < off) {
        float ov = sval[t+off]; int oi = sidx[t+off];
        if (ov > sval[t] || (ov == sval[t] && oi < sidx[t])) { sval[t] = ov; sidx[t] = oi; }
      }
      __syncthreads();
    }
    if (t == 0) { o[s] = sidx[0]; lasti = sidx[0]; }
    __syncthreads();
  }
}

__global__ void pn2_gather_ctr_kernel(const float* __restrict__ pos, const int* __restrict__ idx,
                                      float* __restrict__ ctr, int P, int S) {
  int i = blockIdx.x * blockDim.x + threadIdx.x;
  if (i >= BB * S) return;
  int b = i / S;
  int j = idx[i];
  const float* src = pos + ((size_t)b * P + j) * 3;
  float* d = ctr + (size_t)i * 3;
  d[0] = src[0]; d[1] = src[1]; d[2] = src[2];
}

// ---------------------------------------------------------------- radius grouping
__global__ void pn2_radius_kernel(const float* __restrict__ pos, const float* __restrict__ ctr,
                                  int* __restrict__ nid, int* __restrict__ cnts,
                                  int P, int S, float r2) {
  __shared__ float cd[4][RCAP];
  __shared__ int   ci[4][RCAP];
  __shared__ int   ccnt[4];
  int w    = threadIdx.x >> 5;
  int lane = threadIdx.x & 31;
  int wid  = blockIdx.x * 4 + w;
  bool live = (wid < BB * S);
  if (lane == 0) ccnt[w] = 0;
  __syncthreads();
  int b = live ? (wid / S) : 0;
  float cx = 0.f, cy = 0.f, cz = 0.f;
  if (live) { cx = ctr[(size_t)wid*3+0]; cy = ctr[(size_t)wid*3+1]; cz = ctr[(size_t)wid*3+2]; }
  const float* pb = pos + (size_t)b * P * 3;
  if (live) {
    for (int pi = lane; pi < P; pi += 32) {
      float dx = pb[pi*3+0]-cx, dy = pb[pi*3+1]-cy, dz = pb[pi*3+2]-cz;
      float d2 = dx*dx + dy*dy + dz*dz;
      if (d2 <= r2) {
        int slot = atomicAdd(&ccnt[w], 1);
        if (slot < RCAP) { cd[w][slot] = d2; ci[w][slot] = pi; }
      }
    }
  }
  __syncthreads();
  if (!live) return;
  int n = ccnt[w]; if (n > RCAP) n = RCAP;
  int* nb = nid + (size_t)wid * KN;
  float pv = -INFINITY; int ppi = -1;
  int sel = 0;
  for (int k = 0; k < KN; ++k) {
    float bv = INFINITY; int bpi = 0x7fffffff;
    for (int c = lane; c < n; c += 32) {
      float v = cd[w][c]; int pi = ci[w][c];
      bool after = (v > pv) || (v == pv && pi > ppi);
      if (after && (v < bv || (v == bv && pi < bpi))) { bv = v; bpi = pi; }
    }
    for (int off = 16; off > 0; off >>= 1) {
      float ov = __shfl_xor(bv, off, 32);
      int  opi = __shfl_xor(bpi, off, 32);
      if (ov < bv || (ov == bv && opi < bpi)) { bv = ov; bpi = opi; }
    }
    if (!(bv < INFINITY)) break;
    if (lane == 0) nb[k] = bpi;
    pv = bv; ppi = bpi; sel = k + 1;
  }
  if (lane == 0) cnts[wid] = sel;
}

// ---------------------------------------------------------------- message build (padded stride)
__global__ void pn2_build_msg_kernel(const float* __restrict__ x, const float* __restrict__ pos,
                                     const float* __restrict__ ctr, const int* __restrict__ nid,
                                     const int* __restrict__ cnts, float* __restrict__ msg,
                                     int P, int S, int C, int stride) {
  int e = blockIdx.x * blockDim.x + threadIdx.x;
  if (e >= BB * S * KN) return;
  int g = e >> 6, k = e & 63;
  int b = g / S;
  float* m = msg + (size_t)e * stride;
  if (k < cnts[g]) {
    int j = nid[e];
    const float* xr = x + ((size_t)b * P + j) * C;
    for (int c = 0; c < C; ++c) m[c] = xr[c];
    const float* pr = pos + ((size_t)b * P + j) * 3;
    const float* cr = ctr + (size_t)g * 3;
    m[C+0] = pr[0]-cr[0]; m[C+1] = pr[1]-cr[1]; m[C+2] = pr[2]-cr[2];
  } else {
    for (int c = 0; c < C + 3; ++c) m[c] = 0.f;
  }
  for (int c = C + 3; c < stride; ++c) m[c] = 0.f;
}

// ---------------------------------------------------------------- weight packing
// Wp[(k/2)*NP + n] = {W[k][n], W[k+1][n]} zero-padded to Kp (mult 4) x NP (mult 64).
__global__ void pn2_pack_w_kernel(const float* __restrict__ W, const float* __restrict__ b,
                                  v2f* __restrict__ Wp, float* __restrict__ bp,
                                  int K, int N, int Kp, int NP) {
  int i = blockIdx.x * blockDim.x + threadIdx.x;
  int total = (Kp >> 1) * NP;
  if (i < total) {
    int p = i / NP, n = i % NP;
    int k0 = p * 2;
    v2f v;
    v.x = (k0     < K && n < N) ? W[(size_t)k0 * N + n]       : 0.f;
    v.y = (k0 + 1 < K && n < N) ? W[(size_t)(k0 + 1) * N + n] : 0.f;
    Wp[i] = v;
  }
  if (i < NP) bp[i] = (i < N) ? b[i] : 0.f;
}

// ---------------------------------------------------------------- WMMA GEMM (branch-free core)
// C[M,N] = A[M,Kp(zero-padded)] @ W + bias. One wave per 16x64 tile, 4 accumulators.
__global__ void pn2_gemm_kernel(const float* __restrict__ A, int lda,
                                const v2f* __restrict__ Wp, const float* __restrict__ bp,
                                float* __restrict__ C, int ldc,
                                int M, int N, int Kp, int NP) {
  int wid = blockIdx.x * (blockDim.x >> 5) + (threadIdx.x >> 5);
  int nblk = NP >> 6;
  int total = ((M + 15) >> 4) * nblk;
  if (wid >= total) return;
  int mt = wid / nblk, nb = wid % nblk;
  int lane = threadIdx.x & 31, lg = lane >> 4, li = lane & 15;
  int m0 = mt << 4;
  int ra = m0 + li; if (ra >= M) ra = M - 1;
  const v2f* Ap = (const v2f*)(A + (size_t)ra * lda);
  int cb = (nb << 6) + li;
  v8f acc[4] = {};
  #pragma unroll 4
  for (int k0 = 0; k0 < Kp; k0 += 4) {
    int pr = (k0 >> 1) + lg;
    v2f a = Ap[pr];
    const v2f* wr = Wp + (size_t)pr * NP + cb;
    v2f b0 = wr[0], b1 = wr[16], b2 = wr[32], b3 = wr[48];
    acc[0] = __builtin_amdgcn_wmma_f32_16x16x4_f32(false, a, false, b0, (short)0, acc[0], false, false);
    acc[1] = __builtin_amdgcn_wmma_f32_16x16x4_f32(false, a, false, b1, (short)0, acc[1], false, false);
    acc[2] = __builtin_amdgcn_wmma_f32_16x16x4_f32(false, a, false, b2, (short)0, acc[2], false, false);
    acc[3] = __builtin_amdgcn_wmma_f32_16x16x4_f32(false, a, false, b3, (short)0, acc[3], false, false);
  }
  #pragma unroll
  for (int j = 0; j < 4; ++j) {
    int c = cb + j * 16;
    float bias = bp[c];
    #pragma unroll
    for (int v = 0; v < 8; ++v) {
      int r = m0 + lg * 8 + v;
      if (r < M && c < N) C[(size_t)r * ldc + c] = acc[j][v] + bias;
    }
  }
}

// ---------------------------------------------------------------- WMMA GEMM + masked max-agg
// out[g,:] = max over valid rows of (A[g*M+m,:] @ W + bias). One wave per (group, 64-col block).
__global__ void pn2_gemm_maxagg_kernel(const float* __restrict__ A, int lda,
                                       const v2f* __restrict__ Wp, const float* __restrict__ bp,
                                       float* __restrict__ out, const int* __restrict__ cnts,
                                       int G, int M, int N, int Kp, int NP) {
  int wid = blockIdx.x * (blockDim.x >> 5) + (threadIdx.x >> 5);
  int nblk = NP >> 6;
  if (wid >= G * nblk) return;
  int g = wid / nblk, nb = wid % nblk;
  int lane = threadIdx.x & 31, lg = lane >> 4, li = lane & 15;
  int cb = (nb << 6) + li;
  int lim = cnts ? cnts[g] : M;
  if (lim > M) lim = M;
  float colmax[4] = {-INFINITY, -INFINITY, -INFINITY, -INFINITY};
  for (int m0 = 0; m0 < M; m0 += 16) {
    int ra = m0 + li; if (ra >= M) ra = M - 1;
    const v2f* Ap = (const v2f*)(A + ((size_t)g * M + ra) * lda);
    v8f acc[4] = {};
    #pragma unroll 2
    for (int k0 = 0; k0 < Kp; k0 += 4) {
      int pr = (k0 >> 1) + lg;
      v2f a = Ap[pr];
      const v2f* wr = Wp + (size_t)pr * NP + cb;
      v2f b0 = wr[0], b1 = wr[16], b2 = wr[32], b3 = wr[48];
      acc[0] = __builtin_amdgcn_wmma_f32_16x16x4_f32(false, a, false, b0, (short)0, acc[0], false, false);
      acc[1] = __builtin_amdgcn_wmma_f32_16x16x4_f32(false, a, false, b1, (short)0, acc[1], false, false);
      acc[2] = __builtin_amdgcn_wmma_f32_16x16x4_f32(false, a, false, b2, (short)0, acc[2], false, false);
      acc[3] = __builtin_amdgcn_wmma_f32_16x16x4_f32(false, a, false, b3, (short)0, acc[3], false, false);
    }
    #pragma unroll
    for (int j = 0; j < 4; ++j) {
      #pragma unroll
      for (int v = 0; v < 8; ++v) {
        int row = m0 + lg * 8 + v;
        if (row < lim) colmax[j] = fmaxf(colmax[j], acc[j][v]);
      }
    }
  }
  #pragma unroll
  for (int j = 0; j < 4; ++j) {
    float other = __shfl_xor(colmax[j], 16, 32);
    float m = fmaxf(colmax[j], other);
    int c = cb + j * 16;
    if (lg == 0 && c < N) out[(size_t)g * N + c] = m + bp[c];
  }
}

// ---------------------------------------------------------------- BatchNorm (global, masked)
__global__ void pn2_zero_stats_kernel(float* __restrict__ s, int n) {
  int i = blockIdx.x * blockDim.x + threadIdx.x;
  if (i < n) s[i] = 0.f;
}

__global__ void pn2_bn_stats_kernel(const float* __restrict__ X, float* __restrict__ stats,
                                    const int* __restrict__ cnts, int R, int C) {
  int c = blockIdx.x, t = threadIdx.x;
  float s = 0.f, s2 = 0.f;
  for (int r = t; r < R; r += 256) {
    bool valid = cnts ? ((r & 63) < cnts[r >> 6]) : true;
    if (valid) { float v = X[(size_t)r * C + c]; s += v; s2 += v * v; }
  }
  __shared__ float sa[256], sb[256];
  sa[t] = s; sb[t] = s2;
  __syncthreads();
  for (int off = 128; off > 0; off >>= 1) {
    if (t < off) { sa[t] += sa[t+off]; sb[t] += sb[t+off]; }
    __syncthreads();
  }
  if (t == 0) { stats[c] = sa[0]; stats[C + c] = sb[0]; }
}

__global__ void pn2_bn_count_kernel(float* __restrict__ stats, const int* __restrict__ cnts,
                                    int G, int R, int C) {
  __shared__ int sh[256];
  int t = threadIdx.x, acc = 0;
  if (cnts) for (int i = t; i < G; i += 256) acc += cnts[i];
  sh[t] = acc;
  __syncthreads();
  for (int off = 128; off > 0; off >>= 1) {
    if (t < off) sh[t] += sh[t+off];
    __syncthreads();
  }
  if (t == 0) stats[2 * C] = cnts ? (float)sh[0] : (float)R;
}

__global__ void pn2_bn_norm_relu_kernel(float* __restrict__ X, const float* __restrict__ stats,
                                        const float* __restrict__ g, const float* __restrict__ bt,
                                        int R, int C) {
  size_t i = (size_t)blockIdx.x * blockDim.x + threadIdx.x;
  if (i >= (size_t)R * C) return;
  int c = (int)(i % C);
  float cnt = fmaxf(stats[2 * C], 1.0f);
  float mu  = stats[c] / cnt;
  float var = stats[C + c] / cnt - mu * mu;
  float v = (X[i] - mu) * rsqrtf(var + EPS_BN) * g[c] + bt[c];
  X[i] = fmaxf(v, 0.f);
}

__global__ void pn2_relu_kernel(float* __restrict__ X, size_t n) {
  size_t i = (size_t)blockIdx.x * blockDim.x + threadIdx.x;
  if (i < n) X[i] = fmaxf(X[i], 0.f);
}

// ---------------------------------------------------------------- concat / interpolation (padded)
__global__ void pn2_concat_rows_kernel(const float* __restrict__ a, const float* __restrict__ b,
                                       float* __restrict__ out, int R, int Ca, int Cb, int stride) {
  int i = blockIdx.x * blockDim.x + threadIdx.x;
  if (i >= R) return;
  float* o = out + (size_t)i * stride;
  const float* ar = a + (size_t)i * Ca;
  for (int c = 0; c < Ca; ++c) o[c] = ar[c];
  const float* br = b + (size_t)i * Cb;
  for (int c = 0; c < Cb; ++c) o[Ca + c] = br[c];
  for (int c = Ca + Cb; c < stride; ++c) o[c] = 0.f;
}

__global__ void pn2_concat_bcast_kernel(const float* __restrict__ xg, const float* __restrict__ xl,
                                        float* __restrict__ out, int S, int Cg, int Cl, int stride) {
  int i = blockIdx.x * blockDim.x + threadIdx.x;
  if (i >= BB * S) return;
  int b = i / S;
  float* o = out + (size_t)i * stride;
  const float* gsrc = xg + (size_t)b * Cg;
  for (int c = 0; c < Cg; ++c) o[c] = gsrc[c];
  const float* l = xl + (size_t)i * Cl;
  for (int c = 0; c < Cl; ++c) o[Cg + c] = l[c];
  for (int c = Cg + Cl; c < stride; ++c) o[c] = 0.f;
}

__global__ void pn2_knn3_kernel(const float* __restrict__ psrc, const float* __restrict__ pdst,
                                int* __restrict__ idx3, float* __restrict__ w3, int Ps, int Pd) {
  int i = blockIdx.x * blockDim.x + threadIdx.x;
  if (i >= BB * Pd) return;
  int b = i / Pd;
  float dx0 = pdst[(size_t)i*3+0], dy0 = pdst[(size_t)i*3+1], dz0 = pdst[(size_t)i*3+2];
  const float* sp = psrc + (size_t)b * Ps * 3;
  float d0 = INFINITY, d1 = INFINITY, d2 = INFINITY;
  int i0 = 0, i1 = 0, i2 = 0;
  for (int j = 0; j < Ps; ++j) {
    float ax = sp[j*3+0]-dx0, ay = sp[j*3+1]-dy0, az = sp[j*3+2]-dz0;
    float d = ax*ax + ay*ay + az*az;
    if (d < d0)      { d2=d1; i2=i1; d1=d0; i1=i0; d0=d; i0=j; }
    else if (d < d1) { d2=d1; i2=i1; d1=d;  i1=j; }
    else if (d < d2) { d2=d;  i2=j; }
  }
  idx3[(size_t)i*3+0]=i0; idx3[(size_t)i*3+1]=i1; idx3[(size_t)i*3+2]=i2;
  w3[(size_t)i*3+0]=1.f/fmaxf(d0,1e-16f);
  w3[(size_t)i*3+1]=1.f/fmaxf(d1,1e-16f);
  w3[(size_t)i*3+2]=1.f/fmaxf(d2,1e-16f);
}

__global__ void pn2_interp_concat_kernel(const float* __restrict__ xsrc, const float* __restrict__ xskip,
                                         const int* __restrict__ idx3, const float* __restrict__ w3,
                                         float* __restrict__ out, int Ps, int Pd, int C, int Cs,
                                         int stride) {
  int i = blockIdx.x * blockDim.x + threadIdx.x;
  if (i >= BB * Pd) return;
  int b = i / Pd;
  int j0 = idx3[(size_t)i*3+0], j1 = idx3[(size_t)i*3+1], j2 = idx3[(size_t)i*3+2];
  float w0 = w3[(size_t)i*3+0], w1 = w3[(size_t)i*3+1], w2 = w3[(size_t)i*3+2];
  float inv = 1.f / (w0 + w1 + w2);
  const float* a0 = xsrc + ((size_t)b * Ps + j0) * C;
  const float* a1 = xsrc + ((size_t)b * Ps + j1) * C;
  const float* a2 = xsrc + ((size_t)b * Ps + j2) * C;
  float* o = out + (size_t)i * stride;
  for (int c = 0; c < C; ++c) o[c] = (w0*a0[c] + w1*a1[c] + w2*a2[c]) * inv;
  const float* sk = xskip + (size_t)i * Cs;
  for (int c = 0; c < Cs; ++c) o[C + c] = sk[c];
  for (int c = C + Cs; c < stride; ++c) o[c] = 0.f;
}

// ---------------------------------------------------------------- orchestration
extern "C" void kernel_launch(void* const* d_in, const int* in_sizes, int n_in,
                              void* d_out, int out_size, void* d_ws, size_t ws_size,
                              hipStream_t stream) {
  (void)in_sizes; (void)n_in; (void)out_size; (void)ws_size;
  const float* x0   = (const float*)d_in[0];
  const float* pos0 = (const float*)d_in[1];
  // params flattened in dict order: sa1(3), sa2(3), sa3(3), fp3(2), fp2(2), fp1(3), head(3);
  // each layer = (W, b, gamma, beta)
  auto LW = [&](int L){ return (const float*)d_in[3 + L*4 + 0]; };
  auto LB = [&](int L){ return (const float*)d_in[3 + L*4 + 1]; };
  auto LG = [&](int L){ return (const float*)d_in[3 + L*4 + 2]; };
  auto LT = [&](int L){ return (const float*)d_in[3 + L*4 + 3]; };

  struct LDm { int K, N; };
  static const LDm ld[19] = {
    {6,64},{64,64},{64,128},          // sa1
    {131,128},{128,128},{128,256},    // sa2
    {259,256},{256,512},{512,1024},   // sa3
    {1280,256},{256,256},             // fp3
    {384,256},{256,128},              // fp2
    {131,128},{128,128},{128,128},    // fp1
    {128,128},{128,128},{128,3}       // head
  };
  auto pad4  = [](int k){ return (k + 3) & ~3; };
  auto pad64 = [](int n){ return (n + 63) & ~63; };

  char* wsp = (char*)d_ws;
  auto alloc = [&](size_t bytes) -> void* {
    void* p = (void*)wsp; wsp += (bytes + 255) & ~(size_t)255; return p;
  };
  size_t E1 = (size_t)BB * NS1 * KN, E2 = (size_t)BB * NS2 * KN;
  int*   idx1 = (int*)  alloc((size_t)BB*NS1*sizeof(int));
  int*   idx2 = (int*)  alloc((size_t)BB*NS2*sizeof(int));
  float* ctr1 = (float*)alloc((size_t)BB*NS1*3*sizeof(float));
  float* ctr2 = (float*)alloc((size_t)BB*NS2*3*sizeof(float));
  int*   nid1 = (int*)  alloc(E1*sizeof(int));
  int*   cnt1 = (int*)  alloc((size_t)BB*NS1*sizeof(int));
  int*   nid2 = (int*)  alloc(E2*sizeof(int));
  int*   cnt2 = (int*)  alloc((size_t)BB*NS2*sizeof(int));
  float* x1   = (float*)alloc((size_t)BB*NS1*128*sizeof(float));
  float* x2   = (float*)alloc((size_t)BB*NS2*256*sizeof(float));
  float* x3   = (float*)alloc((size_t)BB*1024*sizeof(float));
  float* f3   = (float*)alloc((size_t)BB*NS2*256*sizeof(float));
  int*   k2i  = (int*)  alloc((size_t)BB*NS1*3*sizeof(int));
  float* k2w  = (float*)alloc((size_t)BB*NS1*3*sizeof(float));
  int*   k1i  = (int*)  alloc((size_t)BB*PP*3*sizeof(int));
  float* k1w  = (float*)alloc((size_t)BB*PP*3*sizeof(float));
  float* stats= (float*)alloc(2112*sizeof(float));

  // packed weights / biases
  size_t wpoff[19], bpoff[19], wptot = 0, bptot = 0;
  for (int L = 0; L < 19; ++L) {
    int Kp = pad4(ld[L].K), NP = pad64(ld[L].N);
    wpoff[L] = wptot; wptot += (size_t)(Kp >> 1) * NP;
    bpoff[L] = bptot; bptot += NP;
  }
  v2f*   WPb = (v2f*)  alloc(wptot * sizeof(v2f));
  float* BPb = (float*)alloc(bptot * sizeof(float));
  float* BUFA = (float*)alloc(E1 * 64 * sizeof(float));   // ping (107 MB)
  float* BUFB = (float*)alloc(E1 * 64 * sizeof(float));   // pong (107 MB)

  for (int L = 0; L < 19; ++L) {
    int Kp = pad4(ld[L].K), NP = pad64(ld[L].N);
    int total = (Kp >> 1) * NP;
    pn2_pack_w_kernel<<<dim3((total + 255)/256), dim3(256), 0, stream>>>(
        LW(L), LB(L), WPb + wpoff[L], BPb + bpoff[L], ld[L].K, ld[L].N, Kp, NP);
  }

  auto gemm = [&](const float* A, int lda, int L, float* C, int ldc, int M) {
    int Kp = pad4(ld[L].K), NP = pad64(ld[L].N);
    int tiles = ((M + 15) / 16) * (NP >> 6);
    pn2_gemm_kernel<<<dim3((tiles + 3) / 4), dim3(128), 0, stream>>>(
        A, lda, WPb + wpoff[L], BPb + bpoff[L], C, ldc, M, ld[L].N, Kp, NP);
  };
  auto bn = [&](float* X, int L, const int* cnts, int G, int R, int C) {
    pn2_zero_stats_kernel<<<dim3((2*C + 1 + 255)/256), dim3(256), 0, stream>>>(stats, 2*C + 1);
    pn2_bn_stats_kernel<<<dim3(C), dim3(256), 0, stream>>>(X, stats, cnts, R, C);
    pn2_bn_count_kernel<<<dim3(1), dim3(256), 0, stream>>>(stats, cnts, G, R, C);
    size_t n = (size_t)R * C;
    pn2_bn_norm_relu_kernel<<<dim3((unsigned)((n + 255)/256)), dim3(256), 0, stream>>>(
        X, stats, LG(L), LT(L), R, C);
  };
  auto maxagg = [&](const float* A, int lda, int L, float* O, const int* cnts, int G, int M) {
    int Kp = pad4(ld[L].K), NP = pad64(ld[L].N);
    int tiles = G * (NP >> 6);
    pn2_gemm_maxagg_kernel<<<dim3((tiles + 3) / 4), dim3(128), 0, stream>>>(
        A, lda, WPb + wpoff[L], BPb + bpoff[L], O, cnts, G, M, ld[L].N, Kp, NP);
  };

  // ---- SA1: 4096 -> 820 centers, r=0.2, MLP [6,64,64,128], masked BN
  pn2_fps_kernel<<<dim3(BB), dim3(256), 0, stream>>>(pos0, idx1, PP, NS1);
  pn2_gather_ctr_kernel<<<dim3((BB*NS1 + 255)/256), dim3(256), 0, stream>>>(pos0, idx1, ctr1, PP, NS1);
  pn2_radius_kernel<<<dim3((BB*NS1 + 3)/4), dim3(128), 0, stream>>>(pos0, ctr1, nid1, cnt1, PP, NS1, 0.04f);
  pn2_build_msg_kernel<<<dim3(((int)E1 + 255)/256), dim3(256), 0, stream>>>(
      x0, pos0, ctr1, nid1, cnt1, BUFA, PP, NS1, 3, 8);
  gemm(BUFA, 8, 0, BUFB, 64, (int)E1);
  bn(BUFB, 0, cnt1, BB*NS1, (int)E1, 64);
  gemm(BUFB, 64, 1, BUFA, 64, (int)E1);
  bn(BUFA, 1, cnt1, BB*NS1, (int)E1, 64);
  maxagg(BUFA, 64, 2, x1, cnt1, BB*NS1, KN);

  // ---- SA2: 820 -> 205 centers, r=0.4, MLP [131,128,128,256]
  pn2_fps_kernel<<<dim3(BB), dim3(256), 0, stream>>>(ctr1, idx2, NS1, NS2);
  pn2_gather_ctr_kernel<<<dim3((BB*NS2 + 255)/256), dim3(256), 0, stream>>>(ctr1, idx2, ctr2, NS1, NS2);
  pn2_radius_kernel<<<dim3((BB*NS2 + 3)/4), dim3(128), 0, stream>>>(ctr1, ctr2, nid2, cnt2, NS1, NS2, 0.16f);
  pn2_build_msg_kernel<<<dim3(((int)E2 + 255)/256), dim3(256), 0, stream>>>(
      x1, ctr1, ctr2, nid2, cnt2, BUFA, NS1, NS2, 128, 132);
  gemm(BUFA, 132, 3, BUFB, 128, (int)E2);
  bn(BUFB, 3, cnt2, BB*NS2, (int)E2, 128);
  gemm(BUFB, 128, 4, BUFA, 128, (int)E2);
  bn(BUFA, 4, cnt2, BB*NS2, (int)E2, 128);
  maxagg(BUFA, 128, 5, x2, cnt2, BB*NS2, KN);

  // ---- SA3 global: MLP [259,256,512,1024] + global max over 205 rows/cloud
  int R3 = BB * NS2;
  pn2_concat_rows_kernel<<<dim3((R3 + 255)/256), dim3(256), 0, stream>>>(x2, ctr2, BUFA, R3, 256, 3, 260);
  gemm(BUFA, 260, 6, BUFB, 256, R3);
  bn(BUFB, 6, nullptr, 0, R3, 256);
  gemm(BUFB, 256, 7, BUFA, 512, R3);
  bn(BUFA, 7, nullptr, 0, R3, 512);
  maxagg(BUFA, 512, 8, x3, nullptr, BB, NS2);

  // ---- FP3 (k=1 interp == broadcast): MLP [1280,256,256]
  pn2_concat_bcast_kernel<<<dim3((R3 + 255)/256), dim3(256), 0, stream>>>(x3, x2, BUFA, NS2, 1024, 256, 1280);
  gemm(BUFA, 1280, 9, BUFB, 256, R3);
  bn(BUFB, 9, nullptr, 0, R3, 256);
  gemm(BUFB, 256, 10, f3, 256, R3);

  // ---- FP2 (k=3): interp f3 p2->p1, concat x1, MLP [384,256,128]
  int R2 = BB * NS1;
  pn2_knn3_kernel<<<dim3((R2 + 255)/256), dim3(256), 0, stream>>>(ctr2, ctr1, k2i, k2w, NS2, NS1);
  pn2_interp_concat_kernel<<<dim3((R2 + 255)/256), dim3(256), 0, stream>>>(
      f3, x1, k2i, k2w, BUFA, NS2, NS1, 256, 128, 384);
  gemm(BUFA, 384, 11, BUFB, 256, R2);
  bn(BUFB, 11, nullptr, 0, R2, 256);
  float* f2 = BUFA;  // reuse ping for f2 [R2,128]
  gemm(BUFB, 256, 12, f2, 128, R2);

  // ---- FP1 (k=3): interp f2 p1->p0, concat x0, MLP [131,128,128,128]
  int R1 = BB * PP;
  pn2_knn3_kernel<<<dim3((R1 + 255)/256), dim3(256), 0, stream>>>(ctr1, pos0, k1i, k1w, NS1, PP);
  float* fin1 = BUFB;  // [R1,132]
  pn2_interp_concat_kernel<<<dim3((R1 + 255)/256), dim3(256), 0, stream>>>(
      f2, x0, k1i, k1w, fin1, NS1, PP, 128, 3, 132);
  float* h1 = BUFA + (size_t)R2 * 128;  // after f2 region (plenty of space)
  gemm(fin1, 132, 13, h1, 128, R1);
  bn(h1, 13, nullptr, 0, R1, 128);
  gemm(h1, 128, 14, BUFB, 128, R1);
  bn(BUFB, 14, nullptr, 0, R1, 128);
  gemm(BUFB, 128, 15, h1, 128, R1);   // f1 (plain last layer)

  // ---- head: [128,128,128,3], norm=False (Lin->ReLU, Lin->ReLU, Lin)
  size_t nh = (size_t)R1 * 128;
  gemm(h1, 128, 16, BUFB, 128, R1);
  pn2_relu_kernel<<<dim3((unsigned)((nh + 255)/256)), dim3(256), 0, stream>>>(BUFB, nh);
  gemm(BUFB, 128, 17, h1, 128, R1);
  pn2_relu_kernel<<<dim3((unsigned)((nh + 255)/256)), dim3(256), 0, stream>>>(h1, nh);
  gemm(h1, 128, 18, (float*)d_out, 3, R1);
}